// Mamba_85418309583282
// MI455X (gfx1250) — compile-verified
//
#include <hip/hip_runtime.h>
#include <hip/hip_bf16.h>

typedef __attribute__((ext_vector_type(2))) float v2f;
typedef __attribute__((ext_vector_type(4))) float v4f;
typedef __attribute__((ext_vector_type(8))) float v8f;

#define BATCH   128
#define T_SEQ   1024
#define DIMS    512
#define CHUNKS  32
#define TSTEPS  (T_SEQ / CHUNKS)   // 32 time steps per workgroup

// ---------------------------------------------------------------------------
// M = A^T  (512x512)
// ---------------------------------------------------------------------------
__global__ void transpose512(const float* __restrict__ A, float* __restrict__ M) {
    int idx = blockIdx.x * blockDim.x + threadIdx.x;   // 0..262143
    int i = idx >> 9;
    int j = idx & 511;
    M[idx] = A[j * 512 + i];
}

// W row 0 = C
__global__ void copy_row512(const float* __restrict__ C, float* __restrict__ W) {
    int d = blockIdx.x * blockDim.x + threadIdx.x;
    if (d < 512) W[d] = C[d];
}

// ---------------------------------------------------------------------------
// D[Mrows,512] = X[Mrows,512] @ Y    (Y is [512,512]; TRANSY -> use Y^T)
// fp32 WMMA 16x16x4. Block = 256 threads = 8 waves; each wave owns a 16x16
// D tile; workgroup tile = 32(M) x 64(N). N=K=512 are compile-time so all
// addressing is pointer increments.
//   GUARD=false: every tile is full (Mrows multiple of 32) -> no cndmask in
//                the loop, unconditional stores, zero EXEC manipulation.
//   GUARD=true : partial M handled branch-free (clamped address + cndmask on
//                value, masked stores); used only for the tiny m<32 steps.
// ---------------------------------------------------------------------------
template <bool TRANSY, bool GUARD>
__global__ void gemm_f32_wmma(const float* __restrict__ X,
                              const float* __restrict__ Y,
                              float* __restrict__ D,
                              int Mrows) {
    constexpr int N = 512;
    constexpr int K = 512;

    const int tid  = threadIdx.x;
    const int wave = tid >> 5;
    const int lane = tid & 31;
    const int tileRow = wave >> 2;          // 0..1
    const int tileCol = wave & 3;           // 0..3
    const int m0 = blockIdx.y * 32 + tileRow * 16;
    const int n0 = blockIdx.x * 64 + tileCol * 16;
    const int half = lane >> 4;             // 0 | 1 (K sub-pair select)
    const int l16  = lane & 15;

    const int am = m0 + l16;                // A-fragment row (M)
    const int bn = n0 + l16;                // B-fragment col (N)
    const bool arow_ok = GUARD ? (am < Mrows) : true;
    const int amc = (GUARD && !arow_ok) ? 0 : am;   // clamped: always valid

    const float* __restrict__ xp = X + (size_t)amc * K + 2 * half;
    const float* __restrict__ yp =
        TRANSY ? (Y + (size_t)bn * K + 2 * half)    // Y stored [N,K]
               : (Y + (size_t)(2 * half) * N + bn); // Y stored [K,N]

    v8f acc = {};
#pragma unroll 4
    for (int k = 0; k < K; k += 4) {
        v2f av = *(const v2f*)xp;           // global_load_b64, unconditional
        v2f a, b;
        if (GUARD) {
            a.x = arow_ok ? av.x : 0.0f;    // v_cndmask, no EXEC games
            a.y = arow_ok ? av.y : 0.0f;
        } else {
            a = av;
        }
        if (TRANSY) {
            v2f bv = *(const v2f*)yp;       // contiguous along K
            b.x = bv.x; b.y = bv.y;
        } else {
            b.x = yp[0];                    // rows k+2*half, k+2*half+1
            b.y = yp[N];
        }
        acc = __builtin_amdgcn_wmma_f32_16x16x4_f32(
            /*neg_a=*/false, a, /*neg_b=*/false, b,
            /*c_mod=*/(short)0, acc, /*reuse_a=*/false, /*reuse_b=*/false);
        xp += 4;
        yp += TRANSY ? 4 : 4 * N;
    }

    // D: VGPR r -> row m0 + r + 8*half, col n0 + l16
    float* __restrict__ dp = D + (size_t)(m0 + 8 * half) * N + n0 + l16;
#pragma unroll
    for (int r = 0; r < 8; ++r) {
        if (GUARD) {
            if (m0 + r + 8 * half < Mrows) dp[(size_t)r * N] = acc[r];
        } else {
            dp[(size_t)r * N] = acc[r];
        }
    }
}

// ---------------------------------------------------------------------------
// Streaming phase: P[b,c] = Σ_{t in chunk c} x[b,t,:] · V[T-1-t,:]
// x (256 MB) read exactly once with non-temporal loads (NT temporal hint) so
// it does not thrash L2; V (2 MB) stays L2-resident, reused by all 128 batches.
// 4096 workgroups x 64 KB keeps HBM at full tilt (~11 us at 23.3 TB/s).
// ---------------------------------------------------------------------------
__global__ void stream_dot(const float* __restrict__ x,
                           const float* __restrict__ V,
                           float* __restrict__ P) {
    const int c = blockIdx.x;               // time chunk
    const int b = blockIdx.y;               // batch
    const int tid = threadIdx.x;            // 256 threads
    const int t0 = c * TSTEPS;

    const v4f* xb = (const v4f*)(x + ((size_t)b * T_SEQ + t0) * DIMS);

    float acc = 0.0f;
    const int n4 = TSTEPS * (DIMS / 4);     // 4096 float4 per workgroup
    for (int f = tid; f < n4; f += 256) {
        const int tl = f >> 7;              // / (DIMS/4)
        const int d4 = f & 127;
        v4f xv = __builtin_nontemporal_load(&xb[f]);
        const v4f* vrow =
            (const v4f*)(V + (size_t)(T_SEQ - 1 - (t0 + tl)) * DIMS);
        v4f vv = vrow[d4];
        acc += xv.x * vv.x + xv.y * vv.y + xv.z * vv.z + xv.w * vv.w;
    }

    __shared__ float s[256];
    s[tid] = acc;
    __syncthreads();
    for (int st = 128; st > 0; st >>= 1) {
        if (tid < st) s[tid] += s[tid + st];
        __syncthreads();
    }
    if (tid == 0) P[b * CHUNKS + c] = s[0];
}

// Deterministic final reduction (no float atomics -> bit-stable across replays)
__global__ void reduce_partials(const float* __restrict__ P,
                                float* __restrict__ y) {
    int b = blockIdx.x * blockDim.x + threadIdx.x;
    if (b < BATCH) {
        float s = 0.0f;
        for (int c = 0; c < CHUNKS; ++c) s += P[b * CHUNKS + c];
        y[b] = s;
    }
}

// ---------------------------------------------------------------------------
extern "C" void kernel_launch(void* const* d_in, const int* in_sizes, int n_in,
                              void* d_out, int out_size, void* d_ws, size_t ws_size,
                              hipStream_t stream) {
    (void)in_sizes; (void)n_in; (void)out_size; (void)ws_size;
    const float* x = (const float*)d_in[0];   // [128,1024,512]
    const float* A = (const float*)d_in[1];   // [512,512]
    const float* B = (const float*)d_in[2];   // [512,512]
    const float* C = (const float*)d_in[3];   // [1,512]
    float* y = (float*)d_out;                 // [128,1]

    char* ws = (char*)d_ws;
    float* W  = (float*)(ws);                           // 1024x512 = 2 MB
    float* V  = (float*)(ws + ((size_t)2 << 20));       // 1024x512 = 2 MB
    float* Mb0 = (float*)(ws + ((size_t)4 << 20));      // 512x512  = 1 MB
    float* Mb1 = (float*)(ws + ((size_t)5 << 20));      // 512x512  = 1 MB
    float* P  = (float*)(ws + ((size_t)6 << 20));       // 4096 floats

    // Phase 1: W[e,:] = C (A^T)^e for e = 0..1023, by log-doubling.
    transpose512<<<1024, 256, 0, stream>>>(A, Mb0);
    copy_row512<<<2, 256, 0, stream>>>(C, W);

    float* Ms[2] = {Mb0, Mb1};
    int cur = 0;
    for (int j = 0; j < 10; ++j) {
        const int m = 1 << j;
        // W[m..2m) = W[0..m) @ M,  M = (A^T)^(2^j)
        dim3 gext(512 / 64, (m + 31) / 32);
        if (m < 32) {   // partial tiles -> guarded variant
            gemm_f32_wmma<false, true><<<gext, 256, 0, stream>>>(
                W, Ms[cur], W + (size_t)m * 512, m);
        } else {        // full tiles -> clean variant
            gemm_f32_wmma<false, false><<<gext, 256, 0, stream>>>(
                W, Ms[cur], W + (size_t)m * 512, m);
        }
        if (j < 9) {                        // M <- M @ M
            dim3 gsq(512 / 64, 512 / 32);
            gemm_f32_wmma<false, false><<<gsq, 256, 0, stream>>>(
                Ms[cur], Ms[cur], Ms[cur ^ 1], 512);
            cur ^= 1;
        }
    }

    // Phase 2: V = W @ B^T   (v_e = B g_e)
    gemm_f32_wmma<true, false><<<dim3(512 / 64, 1024 / 32), 256, 0, stream>>>(
        W, B, V, 1024);

    // Phase 3: stream x once, reduce.
    stream_dot<<<dim3(CHUNKS, BATCH), 256, 0, stream>>>(x, V, P);
    reduce_partials<<<1, 128, 0, stream>>>(P, y);
}